// AvgPoolingMerger_67757404062464
// MI455X (gfx1250) — compile-verified
//
#include <hip/hip_runtime.h>
#include <hip/hip_bf16.h>
#include <stdint.h>

// Problem shape (from reference): B=32, S=2048, D=1536, G=150, K=9, R=32, MAX_TOKENS=150
#define MAX_K   9
#define MAX_D   1536
#define DV4_MAX (MAX_D / 4)   // 384 float4 per row
#define GATHER_THREADS 128    // 4 wave32; 384/128 = 3 column chunks per thread

// ---------------------------------------------------------------------------
// Phase 1: per-sample index resolution, counts, keep-compaction map, attention.
// One block per sample. 256 threads = 8 wave32.
// ---------------------------------------------------------------------------
__global__ __launch_bounds__(256)
void avgmerge_prep(const int* __restrict__ pr,        // [B,2]
                   const int* __restrict__ idx,       // [B,G,K]
                   const int* __restrict__ rem,       // [B,R]
                   int*   __restrict__ gidx_ws,       // [B,G,K] resolved row ids
                   float* __restrict__ invcnt_ws,     // [B,G]
                   int*   __restrict__ srcmap_ws,     // [B,MT] slot -> group (or -1)
                   int*   __restrict__ attn_out,      // [B,MT] int32 output
                   int G, int K, int R, int MT)
{
    const int b   = blockIdx.x;
    const int tid = threadIdx.x;

    __shared__ int rem_s[64];
    __shared__ int keep_s[256];
    __shared__ int srcmap_s[256];

    if (tid < R) rem_s[tid] = rem[(size_t)b * R + tid];
    __syncthreads();

    const int start = pr[2 * b];
    const int end   = pr[2 * b + 1];

    if (tid < G) {
        const int  g  = tid;
        const int* gi = idx     + ((size_t)b * G + g) * K;
        int*       go = gidx_ws + ((size_t)b * G + g) * K;
        int cnt = 0;
        for (int k = 0; k < K; ++k) {
            int  v     = gi[k];
            bool valid = (v != -1);
            if (valid) {
                for (int r = 0; r < R; ++r) {
                    int rv = rem_s[r];
                    if (rv != -1 && rv == v) { valid = false; break; }
                }
            }
            cnt += valid ? 1 : 0;
            // -1 wraps to last row of slice: end + 1 + v
            go[k] = (v >= 0) ? (start + v) : (end + 1 + v);
        }
        keep_s[g] = (cnt > 0) ? 1 : 0;
        // Sum is over ALL K gathered rows; divisor is max(cnt,1).
        invcnt_ws[(size_t)b * G + g] = 1.0f / (float)(cnt > 0 ? cnt : 1);
    }
    if (tid < MT) srcmap_s[tid] = -1;
    __syncthreads();

    if (tid == 0) {
        int nk = 0;
        for (int g = 0; g < G; ++g) nk += keep_s[g];
        int rank = 0;
        for (int g = 0; g < G; ++g) {
            if (keep_s[g]) { srcmap_s[MT - nk + rank] = g; ++rank; }
        }
    }
    __syncthreads();

    if (tid < MT) {
        int sm = srcmap_s[tid];
        srcmap_ws[(size_t)b * MT + tid] = sm;
        attn_out [(size_t)b * MT + tid] = (sm >= 0) ? 1 : 0;
    }
}

// ---------------------------------------------------------------------------
// Phase 2: one block per output row (b, t). Async-stage the K source rows into
// LDS with global_load_async_to_lds_b128, wait on ASYNCcnt, reduce, scale.
// 128 threads = 4 wave32. LDS tile: 9 * 1536 * 4B = 55296 B (< 320KB/WGP).
//
// IOFFSET trick: the async copy adds INST_OFFSET to BOTH the LDS and the
// global address (CDNA5 ISA), and our LDS row layout mirrors
// the global row layout, so one (lds, global) address pair issues 3 copies
// with offset:0 / 2048 / 4096.
// ---------------------------------------------------------------------------
__global__ __launch_bounds__(GATHER_THREADS)
void avgmerge_gather(const float* __restrict__ hidden,   // [B,S,D]
                     const int*   __restrict__ gidx_ws,  // [B,G,K]
                     const float* __restrict__ invcnt_ws,// [B,G]
                     const int*   __restrict__ srcmap_ws,// [B,MT]
                     float*       __restrict__ out,      // [B,MT,D]
                     int S, int D, int G, int K, int MT)
{
    const int slot = blockIdx.x;        // b*MT + t
    const int b    = slot / MT;
    const int tid  = threadIdx.x;
    const int DV4  = D >> 2;

    float4* orow = (float4*)(out + (size_t)slot * D);
    const int g  = srcmap_ws[slot];

    if (g < 0) {  // empty slot: write zeros (d_out is poisoned)
        float4 z = make_float4(0.f, 0.f, 0.f, 0.f);
        for (int c = tid; c < DV4; c += blockDim.x) orow[c] = z;
        return;
    }

    __shared__ float4 sh[MAX_K * DV4_MAX];
    __shared__ int    rows_s[MAX_K];
    if (tid < K) rows_s[tid] = gidx_ws[((size_t)b * G + g) * K + tid];
    __syncthreads();

    const unsigned sh_base = (unsigned)(unsigned long long)(void*)&sh[0];

    if (D == MAX_D && K == MAX_K) {
        // Fast path: 9 rows x 384 float4, 128 threads -> 3 chunks/thread/row.
        #pragma unroll
        for (int k = 0; k < MAX_K; ++k) {
            const unsigned long long gaddr = (unsigned long long)
                ((const char*)(hidden + ((size_t)b * S + rows_s[k]) * MAX_D)
                 + (size_t)tid * 16);
            const unsigned laddr =
                sh_base + (unsigned)(k * DV4_MAX * 16) + (unsigned)tid * 16;
            asm volatile(
                "global_load_async_to_lds_b128 %0, %1, off\n\t"
                "global_load_async_to_lds_b128 %0, %1, off offset:2048\n\t"
                "global_load_async_to_lds_b128 %0, %1, off offset:4096"
                :: "v"(laddr), "v"(gaddr) : "memory");
        }
    } else {
        // Generic fallback.
        const int total = K * DV4;
        for (int i = tid; i < total; i += blockDim.x) {
            int k = i / DV4;
            int c = i - k * DV4;
            const float4* src =
                (const float4*)(hidden + ((size_t)b * S + rows_s[k]) * D) + c;
            unsigned laddr = sh_base + (unsigned)((k * DV4 + c) * 16);
            asm volatile("global_load_async_to_lds_b128 %0, %1, off"
                         :: "v"(laddr), "v"((unsigned long long)src)
                         : "memory");
        }
    }

    // Wait for this wave's async copies, then sync all waves in the block.
    asm volatile("s_wait_asynccnt 0" ::: "memory");
    __syncthreads();

    const float inv = invcnt_ws[(size_t)b * G + g];

    if (D == MAX_D && K == MAX_K) {
        #pragma unroll
        for (int j = 0; j < DV4_MAX / GATHER_THREADS; ++j) {
            const int c = tid + j * GATHER_THREADS;
            float4 acc = sh[c];
            #pragma unroll
            for (int k = 1; k < MAX_K; ++k) {
                float4 v = sh[k * DV4_MAX + c];
                acc.x += v.x; acc.y += v.y; acc.z += v.z; acc.w += v.w;
            }
            acc.x *= inv; acc.y *= inv; acc.z *= inv; acc.w *= inv;
            orow[c] = acc;
        }
    } else {
        for (int c = tid; c < DV4; c += blockDim.x) {
            float4 acc = sh[c];
            for (int k = 1; k < K; ++k) {
                float4 v = sh[k * DV4 + c];
                acc.x += v.x; acc.y += v.y; acc.z += v.z; acc.w += v.w;
            }
            acc.x *= inv; acc.y *= inv; acc.z *= inv; acc.w *= inv;
            orow[c] = acc;
        }
    }
}

// ---------------------------------------------------------------------------
// Host launcher
// ---------------------------------------------------------------------------
extern "C" void kernel_launch(void* const* d_in, const int* in_sizes, int n_in,
                              void* d_out, int out_size, void* d_ws, size_t ws_size,
                              hipStream_t stream)
{
    const float* hidden = (const float*)d_in[0];   // [B,S,D] f32
    // d_in[1] attention_mask, d_in[2] image_grid_thw: unused by the math
    const int* pr  = (const int*)d_in[3];          // [B,2]
    const int* idx = (const int*)d_in[4];          // [B,G,K]
    const int* rem = (const int*)d_in[5];          // [B,R]

    const int B = in_sizes[3] / 2;
    const int S = in_sizes[1] / B;
    const int D = (int)(in_sizes[0] / ((long long)B * S));
    const int R = in_sizes[5] / B;
    const int K = MAX_K;                           // 9 (reference constant)
    const int G = in_sizes[4] / (B * K);
    const int MT = (int)((long long)out_size / ((long long)B * (D + 1)));

    float* out      = (float*)d_out;                         // [B,MT,D] f32
    int*   attn_out = (int*)(out + (size_t)B * MT * D);      // [B,MT] int32

    // Workspace layout
    int*   gidx_ws   = (int*)d_ws;                           // B*G*K ints
    float* invcnt_ws = (float*)(gidx_ws + (size_t)B * G * K);// B*G floats
    int*   srcmap_ws = (int*)(invcnt_ws + (size_t)B * G);    // B*MT ints

    hipLaunchKernelGGL(avgmerge_prep, dim3(B), dim3(256), 0, stream,
                       pr, idx, rem, gidx_ws, invcnt_ws, srcmap_ws, attn_out,
                       G, K, R, MT);

    hipLaunchKernelGGL(avgmerge_gather, dim3(B * MT), dim3(GATHER_THREADS), 0, stream,
                       hidden, gidx_ws, invcnt_ws, srcmap_ws, out,
                       S, D, G, K, MT);
}